// NeuralCache_4793183502803
// MI455X (gfx1250) — compile-verified
//
#include <hip/hip_runtime.h>
#include <hip/hip_bf16.h>
#include <math.h>

typedef float v2f __attribute__((ext_vector_type(2)));
typedef float v8f __attribute__((ext_vector_type(8)));

#define B_ROWS 8192
#define DDIM   512
#define ODIM   512
#define NTAB   4
#define HBITS  16
#define RSIZE  65536
#define TILE_M 16
#define LDS_STRIDE 516   // 512 + 4 pad: WMMA-A column reads hit distinct banks
#define SIM_THRESHOLD 0.3f

__global__ __launch_bounds__(128) void neural_cache_fused(
    const float* __restrict__ x,        // [B, D]
    const float* __restrict__ planes,   // [T, D, H]
    const float* __restrict__ keys,     // [T, R, D] (row-normalized)
    const float* __restrict__ values,   // [T, R, O]
    const unsigned char* __restrict__ valid, // [T, R] bool
    float* __restrict__ out)            // [B*O | B | B | B | B] f32
{
  __shared__ __align__(16) float lds_x[TILE_M * LDS_STRIDE];
  __shared__ unsigned addr_lds[TILE_M][NTAB];

  const int tid  = threadIdx.x;
  const int lane = tid & 31;
  const int wave = tid >> 5;                 // 0..3
  const int row0 = blockIdx.x * TILE_M;

  // ---- stage x tile [16 x 512] -> LDS (padded stride) ----
  {
    const float4* xg = (const float4*)(x + (size_t)row0 * DDIM);
    for (int i = tid; i < TILE_M * (DDIM / 4); i += 128) {
      const int r  = i >> 7;                 // /128 float4 per row
      const int c4 = i & 127;
      *(float4*)&lds_x[r * LDS_STRIDE + c4 * 4] = xg[(size_t)r * (DDIM / 4) + c4];
    }
  }
  __syncthreads();

  // ---- WMMA f32 projection: wave w computes table w's 16x16 proj tile ----
  {
    const float* Wt = planes + (size_t)wave * DDIM * HBITS;
    const int mA = lane & 15;                // A: lane = M (both halves)
    const int kb = (lane >> 4) * 2;          // A/B: K pair {0,1} vs {2,3}
    const int h  = lane & 15;                // B: lane = N (hash bit)
    v8f c = {};
    for (int k0 = 0; k0 < DDIM; k0 += 4) {
      v2f a, bb;
      a.x  = lds_x[mA * LDS_STRIDE + k0 + kb];
      a.y  = lds_x[mA * LDS_STRIDE + k0 + kb + 1];
      bb.x = Wt[(size_t)(k0 + kb)     * HBITS + h];
      bb.y = Wt[(size_t)(k0 + kb + 1) * HBITS + h];
      c = __builtin_amdgcn_wmma_f32_16x16x4_f32(false, a, false, bb,
                                                (short)0, c, false, false);
    }
    // sign bits -> 16-bit LSH addresses: one ballot = 2 complete addresses.
    // Compute all masks first, then ONE lane-0 guarded store block (single
    // exec save/restore instead of 8 divergent regions).
    unsigned masks[8];
#pragma unroll
    for (int v = 0; v < 8; ++v)
      masks[v] = __builtin_amdgcn_ballot_w32(c[v] > 0.0f);
    if (lane == 0) {
#pragma unroll
      for (int v = 0; v < 8; ++v) {
        addr_lds[v][wave]     = masks[v] & 0xFFFFu;  // row v,   bits = lanes 0..15
        addr_lds[v + 8][wave] = masks[v] >> 16;      // row v+8, bits = lanes 16..31
      }
    }
  }
  __syncthreads();

  float* out_vals = out;
  float* out_hit  = out + (size_t)B_ROWS * ODIM;
  float* out_sim  = out_hit  + B_ROWS;
  float* out_addr = out_sim  + B_ROWS;
  float* out_tab  = out_addr + B_ROWS;

  // ---- gather / cosine / select / emit: wave handles rows wave + 4*i ----
  for (int i = 0; i < 4; ++i) {
    const int m = wave + 4 * i;
    const int b = row0 + m;

    // x row -> registers, accumulate |x|^2
    float4 xr[4];
    float ss = 0.f;
#pragma unroll
    for (int it = 0; it < 4; ++it) {
      xr[it] = *(const float4*)&lds_x[m * LDS_STRIDE + it * 128 + lane * 4];
      ss += xr[it].x * xr[it].x + xr[it].y * xr[it].y
          + xr[it].z * xr[it].z + xr[it].w * xr[it].w;
    }
#pragma unroll
    for (int sh = 16; sh >= 1; sh >>= 1) ss += __shfl_xor(ss, sh, 32);
    const float invn = 1.0f / fmaxf(sqrtf(ss), 1e-12f);

    unsigned addrs[NTAB];
#pragma unroll
    for (int t = 0; t < NTAB; ++t) addrs[t] = addr_lds[m][t];

    // issue all 4 gathered key-row loads + valid bytes up front (MLP)
    float4 kr[NTAB][4];
    unsigned char vld[NTAB];
#pragma unroll
    for (int t = 0; t < NTAB; ++t) {
      const float4* kp =
          (const float4*)(keys + ((size_t)t * RSIZE + addrs[t]) * DDIM);
#pragma unroll
      for (int it = 0; it < 4; ++it) kr[t][it] = kp[it * 32 + lane];
      vld[t] = valid[(size_t)t * RSIZE + addrs[t]];
    }

    float best = -INFINITY; int bt = 0; unsigned ba = addrs[0];
#pragma unroll
    for (int t = 0; t < NTAB; ++t) {
      float acc = 0.f;
#pragma unroll
      for (int it = 0; it < 4; ++it) {
        acc += xr[it].x * kr[t][it].x + xr[it].y * kr[t][it].y
             + xr[it].z * kr[t][it].z + xr[it].w * kr[t][it].w;
      }
#pragma unroll
      for (int sh = 16; sh >= 1; sh >>= 1) acc += __shfl_xor(acc, sh, 32);
      const float s  = acc * invn;            // keys pre-normalized: cosine
      const float ms = (vld[t] != 0) ? s : -INFINITY;
      if (ms > best) { best = ms; bt = t; ba = addrs[t]; }  // first-max wins
    }
    const bool hit = best > SIM_THRESHOLD;    // wave-uniform

    if (lane == 0) {
      out_hit[b]  = hit ? 1.0f : 0.0f;
      out_sim[b]  = best;
      out_addr[b] = (float)ba;
      out_tab[b]  = (float)bt;
    }

    // Uniform branch: only touch the 2 KB value row on a hit. With random
    // keys the 0.3 threshold is ~7 sigma, so nearly every row misses and
    // this skips ~16 MB of gathered reads.
    float4* op = (float4*)(out_vals + (size_t)b * ODIM);
    if (hit) {
      const float4* vp =
          (const float4*)(values + ((size_t)bt * RSIZE + ba) * ODIM);
#pragma unroll
      for (int it = 0; it < 4; ++it) op[it * 32 + lane] = vp[it * 32 + lane];
    } else {
      const float4 z = make_float4(0.f, 0.f, 0.f, 0.f);
#pragma unroll
      for (int it = 0; it < 4; ++it) op[it * 32 + lane] = z;
    }
  }
}

extern "C" void kernel_launch(void* const* d_in, const int* in_sizes, int n_in,
                              void* d_out, int out_size, void* d_ws, size_t ws_size,
                              hipStream_t stream) {
  (void)in_sizes; (void)n_in; (void)out_size; (void)d_ws; (void)ws_size;
  const float* x      = (const float*)d_in[0];
  const float* planes = (const float*)d_in[1];
  const float* keys   = (const float*)d_in[2];
  const float* values = (const float*)d_in[3];
  const unsigned char* valid = (const unsigned char*)d_in[4];
  float* out = (float*)d_out;

  dim3 grid(B_ROWS / TILE_M);   // 512 blocks
  dim3 block(128);              // 4 waves: 1 table each for WMMA, 4 rows each after
  hipLaunchKernelGGL(neural_cache_fused, grid, block, 0, stream,
                     x, planes, keys, values, valid, out);
}